// TransportNN_50268297232877
// MI455X (gfx1250) — compile-verified
//
#include <hip/hip_runtime.h>
#include <hip/hip_bf16.h>

#define TEMPERATURE 10.0f
#define Bdim 512
#define Nstar 4096
#define Dfeat 512
#define NCLS 10

typedef __attribute__((ext_vector_type(2))) float v2f;
typedef __attribute__((ext_vector_type(8))) float v8f;

// ---------------- init: zero column-sum accumulators, midx = INT_MAX ----------
__global__ void tn_init(float* csD1, float* csE1, float* csD2, float* csE2, int* midx) {
    int t = blockIdx.x * blockDim.x + threadIdx.x;
    if (t < Bdim) {
        csD1[t] = 0.f; csE1[t] = 0.f; csD2[t] = 0.f; csE2[t] = 0.f;
        midx[t] = 0x7FFFFFFF;
    }
}

// ---------------- row squared-norms: one wave per row ------------------------
__global__ void tn_rownorm(const float* __restrict__ a, float* __restrict__ out,
                           int rows, int K) {
    int wavesPerBlock = blockDim.x >> 5;
    int row  = blockIdx.x * wavesPerBlock + (threadIdx.x >> 5);
    int lane = threadIdx.x & 31;
    if (row >= rows) return;
    float s = 0.f;
    for (int k = lane; k < K; k += 32) {
        float v = a[row * K + k];
        s += v * v;
    }
    #pragma unroll
    for (int off = 16; off > 0; off >>= 1) s += __shfl_down(s, off, 32);
    if (lane == 0) out[row] = s;
}

// ---------------- exact-match scan (first-element filter) --------------------
__global__ void tn_match(const float* __restrict__ x, const float* __restrict__ star,
                         int* __restrict__ midx) {
    int idx = blockIdx.x * blockDim.x + threadIdx.x;   // 512 * 4096
    int b = idx >> 12;
    int n = idx & (Nstar - 1);
    const float* xr = x + (size_t)b * Dfeat;
    const float* sr = star + (size_t)n * Dfeat;
    if (xr[0] != sr[0]) return;                        // prunes ~everything
    for (int k = 1; k < Dfeat; ++k)
        if (xr[k] != sr[k]) return;
    atomicMin(midx + b, n);                            // == argmax of first match
}

// ---------------- squared-distance GEMM: D = max(an + bn - 2*A.B^T, 0) -------
// A [M x K] row-major, Bm [Ncols x K] row-major.
// One wave computes a 16x64 strip: A fragment reused across 4 N-subtiles
// (1 A-load + 4 B-loads per 4 WMMAs -> 1.25 loads/WMMA).
__global__ void tn_sqdist_wmma(const float* __restrict__ A, const float* __restrict__ Bm,
                               const float* __restrict__ an, const float* __restrict__ bn,
                               float* __restrict__ out, float* __restrict__ colsum,
                               int M, int Ncols, int K) {
    int nstrips = Ncols >> 6;               // strips of 64 columns
    int tm = blockIdx.x / nstrips;
    int tn = blockIdx.x % nstrips;
    int lane = threadIdx.x;
    int l16 = lane & 15, hi = lane >> 4;

    const float* arow = A + (size_t)(tm * 16 + l16) * K + 2 * hi;
    const float* brow0 = Bm + (size_t)(tn * 64 +  0 + l16) * K + 2 * hi;
    const float* brow1 = Bm + (size_t)(tn * 64 + 16 + l16) * K + 2 * hi;
    const float* brow2 = Bm + (size_t)(tn * 64 + 32 + l16) * K + 2 * hi;
    const float* brow3 = Bm + (size_t)(tn * 64 + 48 + l16) * K + 2 * hi;

    v8f acc0 = {}, acc1 = {}, acc2 = {}, acc3 = {};
    for (int k = 0; k < K; k += 4) {
        v2f a  = *(const v2f*)(arow  + k);  // A frag: M=l16, K = k + {0,1} (+2 if hi)
        v2f b0 = *(const v2f*)(brow0 + k);  // B frags: N=l16 in each subtile
        v2f b1 = *(const v2f*)(brow1 + k);
        v2f b2 = *(const v2f*)(brow2 + k);
        v2f b3 = *(const v2f*)(brow3 + k);
        acc0 = __builtin_amdgcn_wmma_f32_16x16x4_f32(false, a, false, b0, (short)0, acc0, false, false);
        acc1 = __builtin_amdgcn_wmma_f32_16x16x4_f32(false, a, false, b1, (short)0, acc1, false, false);
        acc2 = __builtin_amdgcn_wmma_f32_16x16x4_f32(false, a, false, b2, (short)0, acc2, false, false);
        acc3 = __builtin_amdgcn_wmma_f32_16x16x4_f32(false, a, false, b3, (short)0, acc3, false, false);
    }

    float anv[8];
    #pragma unroll
    for (int i = 0; i < 8; ++i) anv[i] = an[tm * 16 + i + 8 * hi];

    v8f accs[4] = {acc0, acc1, acc2, acc3};
    #pragma unroll
    for (int j = 0; j < 4; ++j) {
        int n = tn * 64 + j * 16 + l16;
        float bnv = bn[n];
        float csum = 0.f;
        #pragma unroll
        for (int i = 0; i < 8; ++i) {
            int m = tm * 16 + i + 8 * hi;   // C/D layout: VGPR i -> M = i + 8*hi
            float d = anv[i] + bnv - 2.0f * accs[j][i];
            d = fmaxf(d, 0.f);
            out[(size_t)m * Ncols + n] = d;
            csum += d;
        }
        atomicAdd(colsum + n, csum);
    }
}

// ---------------- normalize + exp + column sum -------------------------------
// e = exp(-T * d * rows / colsumD); in-place, accumulate column sums of e.
__global__ void tn_expnorm(float* __restrict__ buf, const float* __restrict__ colsumD,
                           float* __restrict__ colsumE, int rows, int cols,
                           int rowsPerBlock) {
    int c  = blockIdx.x * blockDim.x + threadIdx.x;    // column owner
    int m0 = blockIdx.y * rowsPerBlock;
    float scale = -TEMPERATURE * (float)rows / colsumD[c];
    float s = 0.f;
    for (int m = m0; m < m0 + rowsPerBlock; ++m) {
        float e = __expf(buf[(size_t)m * cols + c] * scale);
        buf[(size_t)m * cols + c] = e;
        s += e;
    }
    atomicAdd(colsumE + c, s);
}

// ---------------- transport GEMM: xt[n,d] = (from^T @ e1)[d,n] / colsumE[n] --
// with exact-match row override. from [K x Dd] row-major, e1 [K x Ncols].
// One wave computes a 16x64 strip (A fragment reused across 4 N-subtiles).
__global__ void tn_transport_wmma(const float* __restrict__ from, const float* __restrict__ e1,
                                  const float* __restrict__ colsumE, const int* __restrict__ midx,
                                  float* __restrict__ xt, int Dd, int Ncols, int K) {
    int nstrips = Ncols >> 6;
    int td = blockIdx.x / nstrips;
    int tn = blockIdx.x % nstrips;
    int lane = threadIdx.x;
    int l16 = lane & 15, hi = lane >> 4;
    int dcol = td * 16 + l16;               // A frag M index (output feature d)

    v8f acc0 = {}, acc1 = {}, acc2 = {}, acc3 = {};
    for (int k = 0; k < K; k += 4) {
        int r0 = k + 2 * hi;
        v2f a;                              // A[M][K] = from[K][M]  (transposed read)
        a.x = from[(size_t)r0 * Dd + dcol];
        a.y = from[(size_t)(r0 + 1) * Dd + dcol];
        const float* er0 = e1 + (size_t)r0 * Ncols + tn * 64 + l16;
        const float* er1 = e1 + (size_t)(r0 + 1) * Ncols + tn * 64 + l16;
        v2f b0, b1, b2, b3;
        b0.x = er0[ 0]; b0.y = er1[ 0];
        b1.x = er0[16]; b1.y = er1[16];
        b2.x = er0[32]; b2.y = er1[32];
        b3.x = er0[48]; b3.y = er1[48];
        acc0 = __builtin_amdgcn_wmma_f32_16x16x4_f32(false, a, false, b0, (short)0, acc0, false, false);
        acc1 = __builtin_amdgcn_wmma_f32_16x16x4_f32(false, a, false, b1, (short)0, acc1, false, false);
        acc2 = __builtin_amdgcn_wmma_f32_16x16x4_f32(false, a, false, b2, (short)0, acc2, false, false);
        acc3 = __builtin_amdgcn_wmma_f32_16x16x4_f32(false, a, false, b3, (short)0, acc3, false, false);
    }

    v8f accs[4] = {acc0, acc1, acc2, acc3};
    #pragma unroll
    for (int j = 0; j < 4; ++j) {
        int ncol = tn * 64 + j * 16 + l16;
        float inv = 1.0f / colsumE[ncol];
        int mi = midx[ncol];
        #pragma unroll
        for (int i = 0; i < 8; ++i) {
            int dd = td * 16 + i + 8 * hi;
            float val = accs[j][i] * inv;
            if (mi < Nstar) val = from[(size_t)mi * Dd + dd];   // matched row override
            xt[(size_t)ncol * Dd + dd] = val;
        }
    }
}

// ---------------- one-hot label weighted average -----------------------------
// y[b,c] = sum_{m: lbl[m]==c} e2[m,b] / colsumE2[b]
__global__ void tn_label_avg(const float* __restrict__ e2, const int* __restrict__ lbl,
                             const float* __restrict__ colsumE2, float* __restrict__ out,
                             int rows, int cols) {
    int b = blockIdx.x * blockDim.x + threadIdx.x;     // column owner
    float acc[NCLS];
    #pragma unroll
    for (int c = 0; c < NCLS; ++c) acc[c] = 0.f;
    for (int m = 0; m < rows; ++m) {
        float e = e2[(size_t)m * cols + b];
        int l = lbl[m];
        #pragma unroll
        for (int c = 0; c < NCLS; ++c) acc[c] += (l == c) ? e : 0.f;
    }
    float inv = 1.0f / colsumE2[b];
    #pragma unroll
    for (int c = 0; c < NCLS; ++c) out[(size_t)b * NCLS + c] = acc[c] * inv;
}

extern "C" void kernel_launch(void* const* d_in, const int* in_sizes, int n_in,
                              void* d_out, int out_size, void* d_ws, size_t ws_size,
                              hipStream_t stream) {
    const float* x    = (const float*)d_in[0];   // [512, 512]
    const float* star = (const float*)d_in[1];   // [4096, 512]
    const float* from = (const float*)d_in[2];   // [4096, 512]
    const int*   lbl  = (const int*)d_in[3];     // [4096]
    // d_in[4..7] = W1,b1,W2,b2: dead code w.r.t. the reference output (y_star).

    float* ws = (float*)d_ws;
    // workspace layout (floats); buf reused for D1/e1 then D2/e2 (~9.5 MB total)
    float* buf    = ws;                  // 4096*512 = 2097152
    float* xt     = ws + 2097152;        // 512*512  =  262144
    float* star_n = ws + 2359296;        // 4096
    float* from_n = ws + 2363392;        // 4096
    float* x_n    = ws + 2367488;        // 512
    float* xt_n   = ws + 2368000;        // 512
    float* csD1   = ws + 2368512;        // 512
    float* csE1   = ws + 2369024;        // 512
    float* csD2   = ws + 2369536;        // 512
    float* csE2   = ws + 2370048;        // 512
    int*   midx   = (int*)(ws + 2370560);// 512
    float* out    = (float*)d_out;       // [512, 10]

    tn_init<<<2, 256, 0, stream>>>(csD1, csE1, csD2, csE2, midx);

    tn_rownorm<<<Nstar / 4, 128, 0, stream>>>(star, star_n, Nstar, Dfeat);
    tn_rownorm<<<Nstar / 4, 128, 0, stream>>>(from, from_n, Nstar, Dfeat);
    tn_rownorm<<<Bdim  / 4, 128, 0, stream>>>(x,    x_n,    Bdim,  Dfeat);

    tn_match<<<(Bdim * Nstar) / 256, 256, 0, stream>>>(x, star, midx);

    // d1 = sqdist(star, x): 256 x 8 strips of 16x64
    tn_sqdist_wmma<<<(Nstar / 16) * (Bdim / 64), 32, 0, stream>>>(
        star, x, star_n, x_n, buf, csD1, Nstar, Bdim, Dfeat);
    tn_expnorm<<<dim3(Bdim / 256, Nstar / 128), 256, 0, stream>>>(
        buf, csD1, csE1, Nstar, Bdim, 128);

    // xt = (from^T @ e1 / colsum)^T with matched override: 32 x 8 strips
    tn_transport_wmma<<<(Dfeat / 16) * (Bdim / 64), 32, 0, stream>>>(
        from, buf, csE1, midx, xt, Dfeat, Bdim, Nstar);
    tn_rownorm<<<Bdim / 4, 128, 0, stream>>>(xt, xt_n, Bdim, Dfeat);

    // d2 = sqdist(from, xt)
    tn_sqdist_wmma<<<(Nstar / 16) * (Bdim / 64), 32, 0, stream>>>(
        from, xt, from_n, xt_n, buf, csD2, Nstar, Bdim, Dfeat);
    tn_expnorm<<<dim3(Bdim / 256, Nstar / 128), 256, 0, stream>>>(
        buf, csD2, csE2, Nstar, Bdim, 128);

    // y_star = (star_labels^T @ e2 / colsum)^T  — segmented sum over labels
    tn_label_avg<<<Bdim / 256, 256, 0, stream>>>(buf, lbl, csE2, out, Nstar, Bdim);
}